// GARCH_6287832121361
// MI455X (gfx1250) — compile-verified
//
#include <hip/hip_runtime.h>

typedef __attribute__((ext_vector_type(2))) float v2f;
typedef __attribute__((ext_vector_type(8))) float v8f;

#define SEGW 2048            // elements per wave-segment
#define TILE 256             // elements per WMMA tile: 16 chunks x 16
#define WPB  8               // waves per block (wave32 -> 256 threads)
#define TPB  (WPB * 32)

__device__ __forceinline__ float powW(float w, int e) {
    float r = 1.f;
    for (int i = 0; i < e; ++i) r *= w;
    return r;
}

// Uniform broadcast from a fixed lane: v_readlane_b32 (no LDS, no DScnt).
__device__ __forceinline__ float readlane_f(float v, int l) {
    return __int_as_float(__builtin_amdgcn_readlane(__float_as_int(v), l));
}

// Pass 1 (STORE=false): per-wave-segment additive aggregate (zero-init scan tail).
// Pass 3 (STORE=true) : full scan output given per-segment entering state.
// Tile math: Y^T = X^T * L^T via 4 chained V_WMMA_F32_16X16X4_F32.
//   X[k][c] = a_{c*16+k},  L[j][k] = Wh^{j-k} (j>=k).
//   acc VGPR v, lane l: holds Y[j=l%16][c = v + 8*(l>>4)].
template <bool STORE>
__global__ __launch_bounds__(TPB) void garch_pass13(
    const float* __restrict__ series, const float* __restrict__ bias_p,
    const float* __restrict__ Wx_p,   const float* __restrict__ Wh_p,
    const float* __restrict__ prefix, // pass3: entering state per segment
    float* __restrict__ out,          // pass3: hist ; pass1: wcarry[nseg]
    int nseg)
{
    const float bias = bias_p[0], Wx = Wx_p[0], Wh = Wh_p[0];
    const int lane = threadIdx.x & 31;
    const int wave = threadIdx.x >> 5;
    const int wid  = blockIdx.x * WPB + wave;
    if (wid >= nseg) return;                 // wave-uniform (EXEC stays all-1s)
    const int n    = lane & 15;              // column index (j)
    const int half = lane >> 4;              // lane half

    // B operand = L^T 4-wide slices (constant). For slice kb, component v in {0,1}
    // holds row K = kb*4 + v + 2*half, column N = n:  B[K][n] = Wh^{n-K} if n>=K.
    v2f bm[4];
#pragma unroll
    for (int kb = 0; kb < 4; ++kb) {
        int kx = kb * 4 + 2 * half;          // component .x row
        int ky = kx + 1;                     // component .y row
        bm[kb].x = (n >= kx) ? powW(Wh, n - kx) : 0.f;
        bm[kb].y = (n >= ky) ? powW(Wh, n - ky) : 0.f;
    }
    const float r16 = powW(Wh, 16);          // per-chunk multiplier
    const float wpj = powW(Wh, n + 1);       // Wh^{j+1} for this lane's j

    const size_t base = (size_t)wid * SEGW;
    float vin = STORE ? prefix[wid] : 0.f;   // entering state of this segment

    for (int tb = 0; tb < SEGW; tb += TILE) {
        const float* sp = series + base + tb;
        v8f acc = {0.f, 0.f, 0.f, 0.f, 0.f, 0.f, 0.f, 0.f};
#pragma unroll
        for (int kb = 0; kb < 4; ++kb) {
            // A operand = X^T slice: lane (m + 16*half) holds row m,
            // K = kb*4 + 2*half + {0,1}  -> contiguous pair, loaded once total.
            v2f s = *(const v2f*)(sp + n * 16 + kb * 4 + 2 * half);
            v2f a;
            a.x = fmaf(Wx, s.x * s.x, bias);
            a.y = fmaf(Wx, s.y * s.y, bias);
            acc = __builtin_amdgcn_wmma_f32_16x16x4_f32(
                false, a, false, bm[kb], (short)0, acc, false, false);
        }
        // Chunk tails Y[15][c]: lane15 holds c=v, lane31 holds c=v+8.
        // v_readlane_b32 -> wave-uniform scalars; no LDS traffic.
        float tails[16];
#pragma unroll
        for (int i = 0; i < 8; ++i) {
            tails[i]     = readlane_f(acc[i], 15);
            tails[i + 8] = readlane_f(acc[i], 31);
        }
        // Serial carry chain across the 16 chunks (uniform, 16 FMAs).
        float u[17];
        u[0] = vin;
#pragma unroll
        for (int c = 0; c < 16; ++c) u[c + 1] = fmaf(r16, u[c], tails[c]);
        vin = u[16];

        if (STORE) {
            float* op = out + base + tb;
#pragma unroll
            for (int v = 0; v < 8; ++v) {
                float uc  = half ? u[v + 8] : u[v];       // entering state of chunk c
                float val = fmaf(wpj, uc, acc[v]);        // Y[j][c] + Wh^{j+1}*u_c
                op[(v + 8 * half) * 16 + n] = val;        // coalesced 64B rows
            }
        }
    }
    if (!STORE) {
        if (lane == 0) out[wid] = vin;       // segment additive aggregate
    }
}

// Pass 2: single block scans nseg segment aggregates with multiplier R = Wh^SEGW,
// folds var0, writes entering state per segment. Weighted Hillis-Steele in LDS.
__global__ __launch_bounds__(256) void garch_pass2(
    const float* __restrict__ wcarry, const float* __restrict__ Wh_p,
    const float* __restrict__ var0_p, float* __restrict__ prefix, int nseg)
{
    const int t = threadIdx.x;
    const float Wh = Wh_p[0];
    float R = powW(Wh, 16);
    for (int i = 16; i < SEGW; i <<= 1) R = R * R;      // R = Wh^SEGW
    const int C = nseg / 256;                            // segments per thread

    // Thread-local inclusive tail over its C segments (zero-init).
    float e = 0.f;
    for (int k = 0; k < C; ++k) e = fmaf(R, e, wcarry[t * C + k]);
    float q = 1.f;
    for (int k = 0; k < C; ++k) q *= R;                  // per-thread multiplier R^C

    __shared__ float sc[256];
    sc[t] = e;
    __syncthreads();
    float m = q;
    for (int d = 1; d < 256; d <<= 1) {
        float add = (t >= d) ? sc[t - d] : 0.f;
        __syncthreads();
        sc[t] = fmaf(m, add, sc[t]);
        m = m * m;
        __syncthreads();
    }
    // Exclusive prefix for this thread + fold initial state var0 (weight q^t).
    float pq = 1.f;
    for (int i = 0; i < t; ++i) pq *= q;
    float P = ((t == 0) ? 0.f : sc[t - 1]) + pq * var0_p[0];
    for (int k = 0; k < C; ++k) {
        prefix[t * C + k] = P;                           // state entering segment
        P = fmaf(R, P, wcarry[t * C + k]);
    }
}

extern "C" void kernel_launch(void* const* d_in, const int* in_sizes, int n_in,
                              void* d_out, int out_size, void* d_ws, size_t ws_size,
                              hipStream_t stream) {
    const float* series = (const float*)d_in[0];
    const float* bias   = (const float*)d_in[1];
    const float* Wx     = (const float*)d_in[2];
    const float* Wh     = (const float*)d_in[3];
    const float* var0   = (const float*)d_in[4];
    float* out = (float*)d_out;

    const int T    = in_sizes[0];        // 8,388,608
    const int nseg = T / SEGW;           // 4096 wave-segments
    float* wcarry = (float*)d_ws;        // [nseg]
    float* prefix = wcarry + nseg;       // [nseg]

    const int blocks = (nseg + WPB - 1) / WPB;   // 512 blocks x 8 waves
    garch_pass13<false><<<blocks, TPB, 0, stream>>>(series, bias, Wx, Wh,
                                                    nullptr, wcarry, nseg);
    garch_pass2<<<1, 256, 0, stream>>>(wcarry, Wh, var0, prefix, nseg);
    garch_pass13<true><<<blocks, TPB, 0, stream>>>(series, bias, Wx, Wh,
                                                   prefix, out, nseg);
}